// CTCLossLayer_5866925326493
// MI455X (gfx1250) — compile-verified
//
#include <hip/hip_runtime.h>
#include <hip/hip_bf16.h>
#include <stdint.h>

// ---------------------------------------------------------------------------
// CTC loss for MI455X (gfx1250, wave32).
//   B=256, T=512, C=128 (blank=127), L=64, S=2L+1=129.
// Two kernels:
//   1) per-(b,t) log-softmax + gather of the 65 referenced classes
//      (blank + 64 labels), written as padded 68-float rows (272B = 17*16B).
//   2) per-batch forward scan: one wave per batch, 4 lattice states per lane
//      (+ s=128 on lane 31) kept in registers; exactly ONE cross-lane shuffle
//      per time step; per-step log-prob row double-buffered in LDS via
//      async global->LDS copies (ASYNCcnt / s_wait_asynccnt).
// Workspace use: B*T*68*4 = 35.7 MB in d_ws.
// ---------------------------------------------------------------------------

#define B_      256
#define T_      512
#define C_      128
#define L_      64
#define S_      129
#define GST     68          // padded gathered-row stride in floats (16B multiple)
#define BLANK_  127
#define NEGF    (-1e30f)

#define AS1 __attribute__((address_space(1)))
#define AS3 __attribute__((address_space(3)))

typedef __attribute__((ext_vector_type(4))) int v4i_t;

#ifndef __has_builtin
#define __has_builtin(x) 0
#endif

#if __has_builtin(__builtin_amdgcn_global_load_async_to_lds_b128)
#define USE_ASYNC_BUILTIN 1
#else
#define USE_ASYNC_BUILTIN 0
#endif

#if __has_builtin(__builtin_amdgcn_s_wait_asynccnt)
#define WAIT_ASYNC(n) __builtin_amdgcn_s_wait_asynccnt(n)
#else
#define WAIT_ASYNC(n) asm volatile("s_wait_asynccnt %0" :: "i"(n) : "memory")
#endif

__device__ __forceinline__ float lae2(float a, float b) {
    float m = fmaxf(a, b);
    return m + __logf(__expf(a - m) + __expf(b - m));
}

__device__ __forceinline__ float lae3(float a, float b, float c) {
    float m = fmaxf(fmaxf(a, b), c);
    return m + __logf(__expf(a - m) + __expf(b - m) + __expf(c - m));
}

// Copy one padded 68-float row (272B) from global to LDS asynchronously.
// 17 lanes issue one b128 each. Completion tracked by ASYNCcnt.
__device__ __forceinline__ void copy_row_async(float* dst, const float* src, int lane) {
    if (lane < 17) {
#if USE_ASYNC_BUILTIN
        AS1 v4i_t* gsrc = (AS1 v4i_t*)(AS1 void*)(void*)(const_cast<float*>(src) + 4 * lane);
        AS3 v4i_t* ldst = (AS3 v4i_t*)(AS3 void*)(void*)(dst + 4 * lane);
        __builtin_amdgcn_global_load_async_to_lds_b128(gsrc, ldst, 0, 0);
#else
        unsigned ldsoff = (unsigned)(size_t)(AS3 void*)(void*)(dst + 4 * lane);
        const float* sp = src + 4 * lane;
        asm volatile("global_load_async_to_lds_b128 %0, %1, off"
                     :: "v"(ldsoff), "v"(sp)
                     : "memory");
#endif
    }
}

// ---------------------------------------------------------------------------
// Kernel 1: per-row logsumexp + gather.
// One wave per (b,t) row; 8 waves (256 threads) per block.
// g[r*GST + 0]      = lp(blank)
// g[r*GST + 1+j]    = lp(y_true[b][j]),  j = 0..63
// ---------------------------------------------------------------------------
extern "C" __global__ void __launch_bounds__(256)
ctc_rowlse_gather(const float* __restrict__ y_pred,
                  const int*   __restrict__ y_true,
                  float*       __restrict__ g)
{
    __shared__ __attribute__((aligned(16))) float rowbuf[8][C_];
    const int lane = threadIdx.x & 31;
    const int w    = threadIdx.x >> 5;
    const int r    = blockIdx.x * 8 + w;   // row id in [0, B*T); exact fit
    const int b    = r >> 9;               // r / T_

    const float4 x = reinterpret_cast<const float4*>(y_pred + (size_t)r * C_)[lane];

    float m = fmaxf(fmaxf(x.x, x.y), fmaxf(x.z, x.w));
#pragma unroll
    for (int off = 16; off; off >>= 1) m = fmaxf(m, __shfl_xor(m, off, 32));

    float s = __expf(x.x - m) + __expf(x.y - m) + __expf(x.z - m) + __expf(x.w - m);
#pragma unroll
    for (int off = 16; off; off >>= 1) s += __shfl_xor(s, off, 32);

    const float lse = m + __logf(s);

    reinterpret_cast<float4*>(&rowbuf[w][0])[lane] = x;
    __syncthreads();

    const int* yb   = y_true + b * L_;
    float*     grow = g + (size_t)r * GST;

    const int c0 = (lane == 0) ? BLANK_ : yb[lane - 1];
    grow[lane]      = rowbuf[w][c0]            - lse;
    grow[lane + 32] = rowbuf[w][yb[lane + 31]] - lse;
    if (lane == 0) grow[64] = rowbuf[w][yb[63]] - lse;
}

// ---------------------------------------------------------------------------
// Kernel 2: forward scan, one wave (32 threads) per batch.
// Lane L owns states s = 4L..4L+3; lane 31 additionally owns s = 128 (a4).
// Even s: new = lae2(alpha[s], alpha[s-1]) + lp_blank
// Odd  s: new = lae3(alpha[s], alpha[s-1], skip ? alpha[s-2] : NEG) + lp_label
// Only cross-lane term needed: alpha[4L-1] = prev lane's a3 (one shfl_up).
// ---------------------------------------------------------------------------
extern "C" __global__ void __launch_bounds__(32)
ctc_scan(const float* __restrict__ g,
         const int*   __restrict__ y_true,
         const int*   __restrict__ input_length,
         const int*   __restrict__ label_length,
         float*       __restrict__ out)
{
    __shared__ __attribute__((aligned(16))) float buf[2][GST];
    __shared__ float afin[S_ + 3];

    const int lane = threadIdx.x;
    const int b    = blockIdx.x;

    const float* gb = g + (size_t)b * T_ * GST;
    int inlen = input_length[b];
    if (inlen > T_) inlen = T_;

    const int* yb = y_true + b * L_;
    // skip masks: s = 2j+1 (j>=1) allows s-2 -> s iff y[j] != y[j-1]
    const int j0  = (lane > 0) ? (2 * lane - 1) : 0;
    const bool sk1 = (lane > 0) && (yb[2 * lane] != yb[j0]);      // k=1: s=4L+1
    const bool sk3 = (yb[2 * lane + 1] != yb[2 * lane]);          // k=3: s=4L+3

    // alpha init at t=0: only s=0 (blank) and s=1 (first label) are live.
    float a0 = (lane == 0) ? gb[0] : NEGF;
    float a1 = (lane == 0) ? gb[1] : NEGF;
    float a2 = NEGF, a3 = NEGF, a4 = NEGF;

    if (inlen > 1) copy_row_async(&buf[1][0], gb + (size_t)1 * GST, lane);

    for (int t = 1; t < inlen; ++t) {
        const bool more = (t + 1) < inlen;
        if (more) {
            copy_row_async(&buf[(t + 1) & 1][0], gb + (size_t)(t + 1) * GST, lane);
            WAIT_ASYNC(1);
        } else {
            WAIT_ASYNC(0);
        }
        asm volatile("" ::: "memory");

        volatile float* row = &buf[t & 1][0];
        const float lblank = row[0];
        const float lo1    = row[2 * lane + 1];   // label lp for s=4L+1
        const float lo2    = row[2 * lane + 2];   // label lp for s=4L+3

        float p3 = __shfl_up(a3, 1, 32);          // alpha[s=4L-1] from lane L-1
        p3 = (lane == 0) ? NEGF : p3;

        const float n0 = lae2(a0, p3)                         + lblank;
        const float n1 = lae3(a1, a0, sk1 ? p3 : NEGF)        + lo1;
        const float n2 = lae2(a2, a1)                         + lblank;
        const float n3 = lae3(a3, a2, sk3 ? a1 : NEGF)        + lo2;
        const float n4 = lae2(a4, a3)                         + lblank;  // s=128 (lane 31)

        a0 = n0; a1 = n1; a2 = n2; a3 = n3; a4 = n4;
    }

    afin[4 * lane + 0] = a0;
    afin[4 * lane + 1] = a1;
    afin[4 * lane + 2] = a2;
    afin[4 * lane + 3] = a3;
    if (lane == 31) afin[128] = a4;
    __syncthreads();

    if (lane == 0) {
        int ll = label_length[b];
        if (ll > L_) ll = L_;
        if (ll < 1)  ll = 1;
        const float ae  = afin[2 * ll];
        const float ae2 = afin[2 * ll - 1];
        const float m   = fmaxf(ae, ae2);
        out[b] = -(m + __logf(__expf(ae - m) + __expf(ae2 - m)));
    }
}

// ---------------------------------------------------------------------------
// Launch: inputs in setup_inputs() order:
//   d_in[0] = y_true        int32  [B, L]
//   d_in[1] = y_pred        f32    [B, T, C]
//   d_in[2] = input_length  int32  [B]
//   d_in[3] = label_length  int32  [B]
// d_out = f32 [B]; d_ws holds the gathered log-prob table (35.7 MB).
// ---------------------------------------------------------------------------
extern "C" void kernel_launch(void* const* d_in, const int* in_sizes, int n_in,
                              void* d_out, int out_size, void* d_ws, size_t ws_size,
                              hipStream_t stream) {
    const int*   y_true  = (const int*)d_in[0];
    const float* y_pred  = (const float*)d_in[1];
    const int*   in_len  = (const int*)d_in[2];
    const int*   lab_len = (const int*)d_in[3];
    float*       outp    = (float*)d_out;
    float*       g       = (float*)d_ws;

    ctc_rowlse_gather<<<(B_ * T_) / 8, 256, 0, stream>>>(y_pred, y_true, g);
    ctc_scan<<<B_, 32, 0, stream>>>(g, y_true, in_len, lab_len, outp);
}